// BaseNode_4432406249711
// MI455X (gfx1250) — compile-verified
//
#include <hip/hip_runtime.h>
#include <stdint.h>

// ---------------------------------------------------------------------------
// IF-neuron scan: y[t] = H(v+x[t] - 1), v = spiked ? 0 : v+x[t]
//   x_seq: [T=64, B*N=262144] fp32, y_seq same shape.
// Memory-bound (128 MB traffic, ~0.4 flop/byte). CDNA5 path used:
// GLOBAL_LOAD_ASYNC_TO_LDS_B128 + S_WAIT_ASYNCCNT depth-8 pipeline, staged in
// the 320KB/WGP LDS so each wave keeps 4KB in flight without VGPR pressure.
// ---------------------------------------------------------------------------

#define AS1 __attribute__((address_space(1)))
#define AS3 __attribute__((address_space(3)))

typedef int   v4i __attribute__((__vector_size__(4 * sizeof(int))));
typedef float v4f __attribute__((ext_vector_type(4)));   // trivially copyable across addr spaces

constexpr int T_STEPS        = 64;
constexpr int DEPTH          = 8;                    // async pipeline depth
constexpr int BLOCK          = 256;                  // 8 waves (wave32)
constexpr int WAVE           = 32;
constexpr int BYTES_PER_LANE = 16;                   // one v4f per lane
constexpr int WAVE_TILE      = WAVE * BYTES_PER_LANE;          // 512 B
constexpr int WAVES_PER_BLK  = BLOCK / WAVE;                   // 8
constexpr int LDS_BYTES      = WAVES_PER_BLK * DEPTH * WAVE_TILE; // 32 KB

__device__ __forceinline__ void async_load_tile(const float* gptr, AS3 uint8_t* lds) {
#if __has_builtin(__builtin_amdgcn_global_load_async_to_lds_b128)
    __builtin_amdgcn_global_load_async_to_lds_b128(
        (AS1 v4i*)gptr, (AS3 v4i*)lds, /*imm offset*/0, /*cpol RT*/0);
#else
    unsigned ldsOff = (unsigned)(uintptr_t)lds;   // AS3 pointers are LDS byte offsets
    asm volatile("global_load_async_to_lds_b128 %0, %1, off"
                 :: "v"(ldsOff), "v"(gptr) : "memory");
#endif
}

template <int N>
__device__ __forceinline__ void wait_asynccnt() {
#if __has_builtin(__builtin_amdgcn_s_wait_asynccnt)
    __builtin_amdgcn_s_wait_asynccnt(N);
#else
    asm volatile("s_wait_asynccnt %0" :: "i"(N));
#endif
}

__global__ __launch_bounds__(BLOCK) void if_neuron_scan_kernel(
    const float* __restrict__ x, float* __restrict__ y, int BN)
{
    __shared__ __align__(16) uint8_t smem[LDS_BYTES];

    const int tid  = threadIdx.x;
    const int lane = tid & (WAVE - 1);
    const int wave = tid >> 5;
    const int idx4 = (blockIdx.x * BLOCK + tid) * 4;   // first of 4 neurons
    if (idx4 + 4 > BN) return;                          // exact fit in practice

    // This lane's private 16B slot inside each of its wave's DEPTH buffers.
    AS3 uint8_t* ldsBase = (AS3 uint8_t*)smem
                         + wave * (DEPTH * WAVE_TILE)
                         + lane * BYTES_PER_LANE;

    const float* gp = x + idx4;   // advances by BN floats per timestep
    float*       yp = y + idx4;

    // ---- prologue: fill the pipeline with tiles t = 0..DEPTH-1 ----
#pragma unroll
    for (int d = 0; d < DEPTH; ++d) {
        async_load_tile(gp, ldsBase + d * WAVE_TILE);
        gp += BN;
    }

    v4f v = {0.f, 0.f, 0.f, 0.f};                 // v_reset = 0

    auto step = [&](int buf) {
        const AS3 v4f* p = (const AS3 v4f*)(ldsBase + buf * WAVE_TILE);
        v4f xv = *p;                               // ds_load_b128
        v4f s;
        v.x += xv.x; s.x = (v.x >= 1.0f) ? 1.0f : 0.0f; v.x = (v.x >= 1.0f) ? 0.0f : v.x;
        v.y += xv.y; s.y = (v.y >= 1.0f) ? 1.0f : 0.0f; v.y = (v.y >= 1.0f) ? 0.0f : v.y;
        v.z += xv.z; s.z = (v.z >= 1.0f) ? 1.0f : 0.0f; v.z = (v.z >= 1.0f) ? 0.0f : v.z;
        v.w += xv.w; s.w = (v.w >= 1.0f) ? 1.0f : 0.0f; v.w = (v.w >= 1.0f) ? 0.0f : v.w;
        *(v4f*)yp = s;                             // global_store_b128 (coalesced 512B/wave)
        yp += BN;
    };

    // ---- steady state: consume tile t, refill with tile t+DEPTH ----
#pragma unroll 8
    for (int t = 0; t < T_STEPS - DEPTH; ++t) {
        wait_asynccnt<DEPTH - 1>();                // oldest tile (t) complete
        const int buf = t & (DEPTH - 1);
        step(buf);                                 // s_wait_dscnt before reuse handled by compiler
        async_load_tile(gp, ldsBase + buf * WAVE_TILE);
        gp += BN;
    }

    // ---- epilogue: drain with shrinking immediate waits (7..0) ----
    // t = T-DEPTH+d  =>  buf = (56+d)&7 = d ; outstanding = DEPTH-d tiles.
#define IF_EPI(d)                                           \
    do {                                                    \
        wait_asynccnt<DEPTH - 1 - (d)>();                   \
        step((T_STEPS - DEPTH + (d)) & (DEPTH - 1));        \
    } while (0)
    IF_EPI(0); IF_EPI(1); IF_EPI(2); IF_EPI(3);
    IF_EPI(4); IF_EPI(5); IF_EPI(6); IF_EPI(7);
#undef IF_EPI
}

extern "C" void kernel_launch(void* const* d_in, const int* in_sizes, int n_in,
                              void* d_out, int out_size, void* d_ws, size_t ws_size,
                              hipStream_t stream)
{
    (void)n_in; (void)out_size; (void)d_ws; (void)ws_size;
    const float* x = (const float*)d_in[0];
    float*       y = (float*)d_out;

    const int total = in_sizes[0];          // T * B * N = 16,777,216
    const int BN    = total / T_STEPS;      // 262,144 neurons

    const int elems_per_block = BLOCK * 4;  // 1024 neurons per block
    const int blocks = (BN + elems_per_block - 1) / elems_per_block;  // 256

    if_neuron_scan_kernel<<<dim3(blocks), dim3(BLOCK), 0, stream>>>(x, y, BN);
}